// MultiheadLeap_23184233464206
// MI455X (gfx1250) — compile-verified
//
#include <hip/hip_runtime.h>
#include <hip/hip_bf16.h>
#include <math.h>

#define S_LEN 4096
#define HID   2048
#define NH    16
#define HD    128
#define WIN   128
#define CHUNK 128
#define EXT   256            // chunk + previous window
#define LSTRIDE 144          // padded LDS row stride (floats); 576B rows keep b128 16B-aligned

typedef float v2f __attribute__((ext_vector_type(2)));
typedef float v4f __attribute__((ext_vector_type(4)));
typedef float v8f __attribute__((ext_vector_type(8)));
typedef int   v4i __attribute__((vector_size(16)));

#define AS_GLOBAL __attribute__((address_space(1)))
#define AS_LDS    __attribute__((address_space(3)))

#if defined(__gfx1250__) && __has_builtin(__builtin_amdgcn_global_load_async_to_lds_b128) && \
    __has_builtin(__builtin_amdgcn_s_wait_asynccnt)
#define USE_ASYNC_LDS 1
#else
#define USE_ASYNC_LDS 0
#endif

__global__ __launch_bounds__(256)
void leap_kernel(const float* __restrict__ qg,
                 const float* __restrict__ f1g,
                 const float* __restrict__ f2g,
                 const float* __restrict__ vg,
                 float* __restrict__ outg)
{
    __shared__ float lds_v[EXT * LSTRIDE];    // 147,456 B: raw v rows; later reused for focused_v
    __shared__ float lds_w[EXT];
    __shared__ float lds_z[CHUNK];

    const int tid  = threadIdx.x;
    const int lane = tid & 31;
    const int wave = tid >> 5;

    const int c  = blockIdx.x & 31;          // seq chunk
    const int h  = (blockIdx.x >> 5) & 15;   // head
    const int b  = blockIdx.x >> 9;          // batch
    const int t0 = c * CHUNK;

    const long sliceBase = (long)b * S_LEN * HID + (long)h * HD;  // row stride = HID floats

    // ---------------- Phase 0: stage raw v block [t0-128, t0+128) into LDS.
    // Async mem->LDS (ASYNCcnt) so the DMA overlaps the phase-1 reductions.
    // u<0 rows: clamp address (finite data); w[u]==0 in the A operand kills them.
#if USE_ASYNC_LDS
    for (int i = 0; i < 32; ++i) {
        const int row = wave * 32 + i;              // uniform within wave
        int u = t0 - WIN + row; u = (u < 0) ? 0 : u;
        const float* src = vg + sliceBase + (long)u * HID + lane * 4;
        __builtin_amdgcn_global_load_async_to_lds_b128(
            (AS_GLOBAL v4i*)src,
            (AS_LDS v4i*)&lds_v[row * LSTRIDE + lane * 4],
            0, 0);
    }
#else
    for (int i = 0; i < 32; ++i) {
        const int e   = i * 256 + tid;     // float4 index over 256x128
        const int row = e >> 5;
        const int c4  = (e & 31) * 4;
        int u = t0 - WIN + row; u = (u < 0) ? 0 : u;
        v4f val = *(const v4f*)(vg + sliceBase + (long)u * HID + c4);
        *(v4f*)(&lds_v[row * LSTRIDE + c4]) = val;
    }
#endif

    // Prefetch this thread's phase-4 q row segment (global_prefetch_b8).
    {
        const int t = tid >> 1, half = tid & 1;
        __builtin_prefetch(qg + sliceBase + (long)(t0 + t) * HID + half * 64, 0, 0);
    }

    // ---------------- Phase 1: focus weights w[ul], ul in [0,256), global u = t0-128+ul
    for (int i = 0; i < 32; ++i) {
        const int ul = wave * 32 + i;
        const int u  = t0 - WIN + ul;
        float s1 = 0.f, s2 = 0.f, q1 = 0.f, q2 = 0.f, xx = 0.f;
        if (u >= 0) {
            const float* r1 = f1g + sliceBase + (long)u * HID + lane * 4;
            const float* r2 = f2g + sliceBase + (long)u * HID + lane * 4;
            v4f a  = *(const v4f*)r1;
            v4f bb = *(const v4f*)r2;
#pragma unroll
            for (int k = 0; k < 4; ++k) {
                float x = a[k], y = bb[k];
                s1 += x; s2 += y; q1 += x * x; q2 += y * y; xx += x * y;
            }
        }
#pragma unroll
        for (int off = 16; off > 0; off >>= 1) {
            s1 += __shfl_xor(s1, off, 32);
            s2 += __shfl_xor(s2, off, 32);
            q1 += __shfl_xor(q1, off, 32);
            q2 += __shfl_xor(q2, off, 32);
            xx += __shfl_xor(xx, off, 32);
        }
        if (lane == 0) {
            float w = 0.f;
            if (u >= 0) {
                const float m1 = s1 * (1.f / HD), m2 = s2 * (1.f / HD);
                const float var1 = (q1 - (float)HD * m1 * m1) * (1.f / (HD - 1));
                const float var2 = (q2 - (float)HD * m2 * m2) * (1.f / (HD - 1));
                const float sd1 = sqrtf(fmaxf(var1, 0.f)) + 1e-5f;
                const float sd2 = sqrtf(fmaxf(var2, 0.f)) + 1e-5f;
                const float dot = (xx - (float)HD * m1 * m2) / (sd1 * sd2);
                w = expf(dot * (5.f / HD));
            }
            lds_w[ul] = w;
        }
    }

#if USE_ASYNC_LDS
    __builtin_amdgcn_s_wait_asynccnt(0);   // this wave's LDS writes landed
#endif
    __syncthreads();                        // w + v block visible workgroup-wide

    // ---------------- z_den[t] = sum_{u=t-127..t} w  (local: ul in [tl+1, tl+128])
    if (tid < CHUNK) {
        float z = 0.f;
#pragma unroll 4
        for (int j = 1; j <= WIN; ++j) z += lds_w[tid + j];
        lds_z[tid] = z;
    }
    __syncthreads();

    // ---------------- Phase 3: banded matmul via V_WMMA_F32_16X16X4_F32
    // s_num(128x128) = A(128x256) x v(256x128), A[t,u] = w[u] * 1[t-127 <= u <= t]
    const int ti = wave;
    const int m  = lane & 15;
    const int hi = lane >> 4;              // 0/1: K-pair select per ISA f32 A/B layout
    const int tl = ti * 16 + m;            // output row within chunk

    // A operands are d-tile invariant: precompute all 36 masked (wx,wy) pairs once.
    // Band mask is all-ones for uo in [1,7]; only the leading (uo=0) and trailing
    // (uo=8) triangles need masking -> multiply by computed 0/1 (no select-fed loads,
    // no divergent branches; EXEC stays all-ones around every WMMA).
    v2f Apre[36];
#pragma unroll
    for (int uo = 0; uo < 9; ++uo) {
#pragma unroll
        for (int ks = 0; ks < 4; ++ks) {
            const int ul0 = (ti + uo) * 16 + ks * 4 + hi * 2;
            float wx = lds_w[ul0];
            float wy = lds_w[ul0 + 1];
            if (uo == 0) {           // lower edge: need ul > tl (upper bound auto-true)
                wx *= (ul0     > tl) ? 1.f : 0.f;
                wy *= (ul0 + 1 > tl) ? 1.f : 0.f;
            } else if (uo == 8) {    // upper edge: need ul <= tl+WIN (lower bound auto-true)
                wx *= (ul0     <= tl + WIN) ? 1.f : 0.f;
                wy *= (ul0 + 1 <= tl + WIN) ? 1.f : 0.f;
            }
            v2f a; a.x = wx; a.y = wy;
            Apre[uo * 4 + ks] = a;
        }
    }

    v8f facc[8];

#pragma unroll
    for (int dj = 0; dj < 8; ++dj) {
        v8f acc = {0.f, 0.f, 0.f, 0.f, 0.f, 0.f, 0.f, 0.f};
#pragma unroll
        for (int uo = 0; uo < 9; ++uo) {
#pragma unroll
            for (int ks = 0; ks < 4; ++ks) {
                const int ul0 = (ti + uo) * 16 + ks * 4 + hi * 2;
                v2f B;
                B.x = lds_v[ul0       * LSTRIDE + dj * 16 + m];
                B.y = lds_v[(ul0 + 1) * LSTRIDE + dj * 16 + m];
                acc = __builtin_amdgcn_wmma_f32_16x16x4_f32(
                    false, Apre[uo * 4 + ks], false, B, (short)0, acc, false, false);
            }
        }
        // C layout: vgpr r, lane -> t = 16*ti + r + 8*hi, d = dj*16 + m
#pragma unroll
        for (int r = 0; r < 8; ++r) {
            const int trow = ti * 16 + r + 8 * hi;
            facc[dj][r] = acc[r] / lds_z[trow];
        }
    }
    __syncthreads();   // all WMMA done -> v region free

    // park focused_v in LDS rows 0..127 (reuse v region)
#pragma unroll
    for (int dj = 0; dj < 8; ++dj) {
#pragma unroll
        for (int r = 0; r < 8; ++r) {
            const int trow = ti * 16 + r + 8 * hi;
            lds_v[trow * LSTRIDE + dj * 16 + m] = facc[dj][r];
        }
    }
    __syncthreads();

    // ---------------- Phase 4: sigmoid(q . fv / sqrt(128)) gate + store
    {
        const int t    = tid >> 1;
        const int half = tid & 1;
        const int dbase = half * 64;
        const long rowOff = sliceBase + (long)(t0 + t) * HID;
        const float* qr = qg + rowOff + dbase;
        float dot = 0.f;
        v4f fv[16];
#pragma unroll
        for (int i = 0; i < 16; ++i) {
            v4f fq = *(const v4f*)(qr + i * 4);
            v4f ff = *(const v4f*)(&lds_v[t * LSTRIDE + dbase + i * 4]);
            fv[i] = ff;
            dot += fq.x * ff.x + fq.y * ff.y + fq.z * ff.z + fq.w * ff.w;
        }
        dot += __shfl_xor(dot, 1, 32);     // combine the two half-rows
        const float align = 1.f / (1.f + expf(-dot * 0.08838834764831845f)); // 1/sqrt(128)
        float* orow = outg + rowOff + dbase;
#pragma unroll
        for (int i = 0; i < 16; ++i) {
            v4f o = fv[i];
            o.x *= align; o.y *= align; o.z *= align; o.w *= align;
            *(v4f*)(orow + i * 4) = o;
        }
    }
}

extern "C" void kernel_launch(void* const* d_in, const int* in_sizes, int n_in,
                              void* d_out, int out_size, void* d_ws, size_t ws_size,
                              hipStream_t stream) {
    (void)in_sizes; (void)n_in; (void)d_ws; (void)ws_size; (void)out_size;
    const float* q  = (const float*)d_in[0];
    const float* f1 = (const float*)d_in[1];
    const float* f2 = (const float*)d_in[2];
    const float* v  = (const float*)d_in[3];
    float* out = (float*)d_out;

    const int nblocks = 4 * NH * (S_LEN / CHUNK);   // 2048
    leap_kernel<<<nblocks, 256, 0, stream>>>(q, f1, f2, v, out);
}